// GNNPolicy_87582973100067
// MI455X (gfx1250) — compile-verified
//
#include <hip/hip_runtime.h>
#include <math.h>

#define NNODES 100000
#define NEDGES 1600000
#define NEP    1700000   // edges + self loops
#define FIN    16
#define FH     64
#define FOUT   8

typedef __attribute__((ext_vector_type(2))) float v2f;
typedef __attribute__((ext_vector_type(8))) float v8f;

__device__ __forceinline__ float leaky(float e) { return e > 0.f ? e : 0.2f * e; }

// ---------------------------------------------------------------------------
// h[N,64] = in[N,K] @ W[K,64]  via V_WMMA_F32_16X16X4_F32 (full fp32 path).
// One wave -> one 16x16 tile. gw uniform per wave => EXEC all-ones for WMMA.
// A layout (ISA 7.12.2, 32-bit A 16x4): lane m=lo, holds K = 2*hi, 2*hi+1.
// B layout (4x16, row striped across lanes): lane col=lo, holds rows 2*hi, 2*hi+1.
// C/D: VGPR r -> row r + 8*hi, col lo.
// ---------------------------------------------------------------------------
__global__ void gemm_h_kernel(const float* __restrict__ in, const float* __restrict__ W,
                              float* __restrict__ h, int K, int nRowTiles) {
    const int lane = threadIdx.x & 31;
    const int gw   = blockIdx.x * 8 + (threadIdx.x >> 5);
    const int rowTile = gw >> 2;   // 4 col tiles cover 64 outputs
    const int colTile = gw & 3;
    if (rowTile >= nRowTiles) return;
    const int hi  = lane >> 4;
    const int lo  = lane & 15;
    const int row = rowTile * 16 + lo;
    const int col = colTile * 16 + lo;
    v8f c = {};
    for (int kk = 0; kk < K; kk += 4) {
        const int k0 = kk + 2 * hi;
        v2f a, b;
        a.x = in[(size_t)row * K + k0];
        a.y = in[(size_t)row * K + k0 + 1];
        b.x = W[(size_t)k0 * FH + col];
        b.y = W[(size_t)(k0 + 1) * FH + col];
        c = __builtin_amdgcn_wmma_f32_16x16x4_f32(false, a, false, b, (short)0, c, false, false);
    }
    const int rbase = rowTile * 16 + 8 * hi;
#pragma unroll
    for (int r = 0; r < 8; ++r)
        h[(size_t)(rbase + r) * FH + col] = c[r];
}

// ---------------------------------------------------------------------------
// Per-node: s_src[n] = h[n]·a_src, s_dst[n] = h[n]·a_dst (wave32 shuffle tree),
// fused with init of m = -inf (bits), den = 0, out = 0.
// ---------------------------------------------------------------------------
__global__ void scalars_init_kernel(const float* __restrict__ h,
                                    const float* __restrict__ a_src,
                                    const float* __restrict__ a_dst,
                                    float* __restrict__ s_src, float* __restrict__ s_dst,
                                    int* __restrict__ mbits, float* __restrict__ den,
                                    float* __restrict__ obuf) {
    const int lane = threadIdx.x & 31;
    const int n    = blockIdx.x * 8 + (threadIdx.x >> 5);
    if (n >= NNODES) return;
    const float h0 = h[(size_t)n * FH + lane];
    const float h1 = h[(size_t)n * FH + lane + 32];
    float ps = h0 * a_src[lane] + h1 * a_src[lane + 32];
    float pd = h0 * a_dst[lane] + h1 * a_dst[lane + 32];
    for (int off = 16; off; off >>= 1) {
        ps += __shfl_xor(ps, off, 32);
        pd += __shfl_xor(pd, off, 32);
    }
    if (lane == 0) {
        s_src[n] = ps;
        s_dst[n] = pd;
        mbits[n] = (int)0xFF800000u;   // -inf
        den[n]   = 0.f;
    }
    obuf[(size_t)n * FH + lane]      = 0.f;
    obuf[(size_t)n * FH + lane + 32] = 0.f;
}

__device__ __forceinline__ void edge_sd(int e, const int* __restrict__ src,
                                        const int* __restrict__ dst, int& s, int& d) {
    if (e < NEDGES) { s = src[e]; d = dst[e]; }
    else            { s = d = e - NEDGES; }      // self loop
}

// segment max of leaky logits via int/uint atomic trick
__global__ void edge_max_kernel(const int* __restrict__ src, const int* __restrict__ dst,
                                const float* __restrict__ s_src, const float* __restrict__ s_dst,
                                int* __restrict__ mbits) {
    const int e = blockIdx.x * blockDim.x + threadIdx.x;
    if (e >= NEP) return;
    int s, d; edge_sd(e, src, dst, s, d);
    const float l = leaky(s_src[s] + s_dst[d]);
    const unsigned u = __float_as_uint(l);
    if ((u >> 31) == 0) atomicMax((int*)&mbits[d], (int)u);
    else                atomicMin((unsigned int*)&mbits[d], u);
}

// segment sum of exp(l - m[dst])
__global__ void edge_den_kernel(const int* __restrict__ src, const int* __restrict__ dst,
                                const float* __restrict__ s_src, const float* __restrict__ s_dst,
                                const int* __restrict__ mbits, float* __restrict__ den) {
    const int e = blockIdx.x * blockDim.x + threadIdx.x;
    if (e >= NEP) return;
    int s, d; edge_sd(e, src, dst, s, d);
    const float l  = leaky(s_src[s] + s_dst[d]);
    const float mv = __int_as_float(mbits[d]);
    atomicAdd(&den[d], expf(l - mv));
}

// heavy pass: one wave per edge; lane = channel; (s,d) wave-uniform -> scalar
// loads for the 4 attention scalars, coalesced 256B gather + 64 f32 atomics.
__global__ void edge_aggr_kernel(const int* __restrict__ src, const int* __restrict__ dst,
                                 const float* __restrict__ s_src, const float* __restrict__ s_dst,
                                 const int* __restrict__ mbits, const float* __restrict__ den,
                                 const float* __restrict__ h, float* __restrict__ out) {
    const int lane = threadIdx.x & 31;
    const int e    = blockIdx.x * 8 + (threadIdx.x >> 5);
    if (e >= NEP) return;
    int s, d; edge_sd(e, src, dst, s, d);
    const float l     = leaky(s_src[s] + s_dst[d]);
    const float mv    = __int_as_float(mbits[d]);
    const float alpha = expf(l - mv) / (den[d] + 1e-16f);
    const float v0 = h[(size_t)s * FH + lane]      * alpha;
    const float v1 = h[(size_t)s * FH + lane + 32] * alpha;
    atomicAdd(&out[(size_t)d * FH + lane],      v0);
    atomicAdd(&out[(size_t)d * FH + lane + 32], v1);
}

// out = relu(out + b) in place
__global__ void act_kernel(float* __restrict__ buf, const float* __restrict__ b) {
    const size_t i = (size_t)blockIdx.x * blockDim.x + threadIdx.x;
    if (i >= (size_t)NNODES * FH) return;
    const float v = buf[i] + b[i & 63];
    buf[i] = v > 0.f ? v : 0.f;
}

// policy head: [N,64]@[64,8]+bp via WMMA, cols 8..15 zero-padded/guarded.
__global__ void policy_kernel(const float* __restrict__ act, const float* __restrict__ Wp,
                              const float* __restrict__ bp, float* __restrict__ out,
                              int nRowTiles) {
    const int lane = threadIdx.x & 31;
    const int gw   = blockIdx.x * 8 + (threadIdx.x >> 5);
    if (gw >= nRowTiles) return;
    const int hi  = lane >> 4;
    const int lo  = lane & 15;
    const int row = gw * 16 + lo;
    v8f c = {};
    for (int kk = 0; kk < FH; kk += 4) {
        const int k0 = kk + 2 * hi;
        v2f a, b;
        a.x = act[(size_t)row * FH + k0];
        a.y = act[(size_t)row * FH + k0 + 1];
        b.x = (lo < FOUT) ? Wp[(size_t)k0 * FOUT + lo]       : 0.f;
        b.y = (lo < FOUT) ? Wp[(size_t)(k0 + 1) * FOUT + lo] : 0.f;
        c = __builtin_amdgcn_wmma_f32_16x16x4_f32(false, a, false, b, (short)0, c, false, false);
    }
    if (lo < FOUT) {
        const int rbase = gw * 16 + 8 * hi;
        const float bb = bp[lo];
#pragma unroll
        for (int r = 0; r < 8; ++r)
            out[(size_t)(rbase + r) * FOUT + lo] = c[r] + bb;
    }
}

extern "C" void kernel_launch(void* const* d_in, const int* in_sizes, int n_in,
                              void* d_out, int out_size, void* d_ws, size_t ws_size,
                              hipStream_t stream) {
    const float* x   = (const float*)d_in[0];
    const int*   ei  = (const int*)  d_in[1];
    const float* W1  = (const float*)d_in[2];
    const float* as1 = (const float*)d_in[3];
    const float* ad1 = (const float*)d_in[4];
    const float* b1  = (const float*)d_in[5];
    const float* W2  = (const float*)d_in[6];
    const float* as2 = (const float*)d_in[7];
    const float* ad2 = (const float*)d_in[8];
    const float* b2  = (const float*)d_in[9];
    const float* Wp  = (const float*)d_in[10];
    const float* bp  = (const float*)d_in[11];
    float* out = (float*)d_out;

    const int* srcA = ei;            // edge_index[0]
    const int* dstA = ei + NEDGES;   // edge_index[1]

    // workspace: h[N*64] | obuf[N*64] | s_src[N] | s_dst[N] | mbits[N] | den[N]
    float* ws   = (float*)d_ws;
    float* hbuf = ws;
    float* obuf = ws + (size_t)NNODES * FH;
    float* ssrc = ws + (size_t)2 * NNODES * FH;
    float* sdst = ssrc + NNODES;
    int*   mb   = (int*)(sdst + NNODES);
    float* den  = (float*)(mb + NNODES);

    const int nRowTiles = NNODES / 16;               // 6250 exact
    const dim3 blk(256);
    const int gemmBlocks = (nRowTiles * 4 + 7) / 8;  // 3125
    const int nodeBlocks = (NNODES + 7) / 8;         // 12500 (wave per node)
    const int edgeBlocks = (NEP + 255) / 256;
    const int aggrBlocks = (NEP + 7) / 8;            // wave per edge
    const int actBlocks  = (NNODES * FH + 255) / 256;
    const int polBlocks  = (nRowTiles + 7) / 8;

    // ---- layer 1 ----
    gemm_h_kernel<<<gemmBlocks, blk, 0, stream>>>(x, W1, hbuf, FIN, nRowTiles);
    scalars_init_kernel<<<nodeBlocks, blk, 0, stream>>>(hbuf, as1, ad1, ssrc, sdst, mb, den, obuf);
    edge_max_kernel<<<edgeBlocks, blk, 0, stream>>>(srcA, dstA, ssrc, sdst, mb);
    edge_den_kernel<<<edgeBlocks, blk, 0, stream>>>(srcA, dstA, ssrc, sdst, mb, den);
    edge_aggr_kernel<<<aggrBlocks, blk, 0, stream>>>(srcA, dstA, ssrc, sdst, mb, den, hbuf, obuf);
    act_kernel<<<actBlocks, blk, 0, stream>>>(obuf, b1);

    // ---- layer 2 (obuf holds act1; gemm consumes it, then it is recycled) ----
    gemm_h_kernel<<<gemmBlocks, blk, 0, stream>>>(obuf, W2, hbuf, FH, nRowTiles);
    scalars_init_kernel<<<nodeBlocks, blk, 0, stream>>>(hbuf, as2, ad2, ssrc, sdst, mb, den, obuf);
    edge_max_kernel<<<edgeBlocks, blk, 0, stream>>>(srcA, dstA, ssrc, sdst, mb);
    edge_den_kernel<<<edgeBlocks, blk, 0, stream>>>(srcA, dstA, ssrc, sdst, mb, den);
    edge_aggr_kernel<<<aggrBlocks, blk, 0, stream>>>(srcA, dstA, ssrc, sdst, mb, den, hbuf, obuf);
    act_kernel<<<actBlocks, blk, 0, stream>>>(obuf, b2);

    // ---- policy head ----
    policy_kernel<<<polBlocks, blk, 0, stream>>>(obuf, Wp, bp, out, nRowTiles);
}